// NonMaxSuppression_66932770341559
// MI455X (gfx1250) — compile-verified
//
#include <hip/hip_runtime.h>
#include <stdint.h>

#define HH 4096
#define WW 4096
#define MASK_N (HH * WW)
#define MAX_KPTS (1 << 20)
#define REL_THR 0.7f
#define REP_THR 0.7f

#if defined(__has_builtin)
#if __has_builtin(__builtin_amdgcn_global_load_async_to_lds_b128) && \
    __has_builtin(__builtin_amdgcn_s_wait_asynccnt)
#define USE_ASYNC_LDS 1
#endif
#if __has_builtin(__builtin_amdgcn_ballot_w32)
#define HAVE_BALLOT_W32 1
#endif
#endif

typedef int v4i __attribute__((ext_vector_type(4)));

// ---------------------------------------------------------------------------
// Pass 1 (placed first so the disasm snippet shows the async-to-LDS path):
// 3x3 max-filter NMS mask, one block (256 threads, 8 waves) per row.
// Rows y-1, y, y+1 staged in LDS (48KB of 320KB/WGP) via
// GLOBAL_LOAD_ASYNC_TO_LDS_B128, completion via s_wait_asynccnt.
// ---------------------------------------------------------------------------
__global__ __launch_bounds__(256) void nms_mask_kernel(
    const float* __restrict__ rel, const float* __restrict__ rep,
    float* __restrict__ mask_out, unsigned* __restrict__ row_counts) {
    __shared__ float rm1[WW];
    __shared__ float r0[WW];
    __shared__ float rp1[WW];
    __shared__ unsigned cnt;

    const int y = blockIdx.x;
    const int tid = threadIdx.x;
    const int ym1 = (y == 0) ? 0 : y - 1;          // clamp == -inf pad for max
    const int yp1 = (y == HH - 1) ? HH - 1 : y + 1;

    const float* gm1 = rep + (size_t)ym1 * WW;
    const float* g0  = rep + (size_t)y   * WW;
    const float* gp1 = rep + (size_t)yp1 * WW;

    if (tid == 0) cnt = 0;

#ifdef USE_ASYNC_LDS
    // 16 bytes/lane * 256 lanes = 4KB per op; 4 ops cover one 16KB row.
#pragma unroll
    for (int i = 0; i < 4; ++i) {
        const int off = (i * 256 + tid) * 4;  // float index, 16B aligned per lane
        __builtin_amdgcn_global_load_async_to_lds_b128(
            (v4i*)(gm1 + off), (v4i*)&rm1[off], 0, 0);
        __builtin_amdgcn_global_load_async_to_lds_b128(
            (v4i*)(g0 + off), (v4i*)&r0[off], 0, 0);
        __builtin_amdgcn_global_load_async_to_lds_b128(
            (v4i*)(gp1 + off), (v4i*)&rp1[off], 0, 0);
    }
    __builtin_amdgcn_s_wait_asynccnt(0);
    __syncthreads();
#else
#pragma unroll
    for (int i = 0; i < 4; ++i) {
        const int off = (i * 256 + tid) * 4;
        *(float4*)&rm1[off] = *(const float4*)&gm1[off];
        *(float4*)&r0[off]  = *(const float4*)&g0[off];
        *(float4*)&rp1[off] = *(const float4*)&gp1[off];
    }
    __syncthreads();
#endif

    unsigned my = 0;
    for (int c = 0; c < 16; ++c) {
        const int x = c * 256 + tid;
        const int xm = (x == 0) ? 0 : x - 1;        // clamp == -inf pad for max
        const int xp = (x == WW - 1) ? WW - 1 : x + 1;
        float a = fmaxf(fmaxf(rm1[xm], rm1[x]), rm1[xp]);
        float b = fmaxf(fmaxf(r0[xm],  r0[x]),  r0[xp]);
        float d = fmaxf(fmaxf(rp1[xm], rp1[x]), rp1[xp]);
        float m = fmaxf(fmaxf(a, b), d);
        float cv = r0[x];
        float rl = rel[(size_t)y * WW + x];
        bool is_kp = (cv == m) && (cv >= REP_THR) && (rl >= REL_THR);
        mask_out[(size_t)y * WW + x] = is_kp ? 1.0f : 0.0f;
        my += is_kp ? 1u : 0u;
    }
    if (my) atomicAdd(&cnt, my);
    __syncthreads();
    if (tid == 0) row_counts[y] = cnt;
}

// ---------------------------------------------------------------------------
// Pass 0: fill coords region with -1.0f (jnp.nonzero fill_value)
// ---------------------------------------------------------------------------
__global__ __launch_bounds__(256) void nms_fill_kernel(float* __restrict__ p, int n, float v) {
    int i = blockIdx.x * blockDim.x + threadIdx.x;
    if (i < n) p[i] = v;
}

// ---------------------------------------------------------------------------
// Pass 2: exclusive prefix sum over 4096 per-row counts (single block).
// ---------------------------------------------------------------------------
__global__ __launch_bounds__(256) void nms_scan_kernel(
    const unsigned* __restrict__ counts, unsigned* __restrict__ rowstart) {
    __shared__ unsigned tsum[256];
    const int tid = threadIdx.x;
    unsigned local[16];
    unsigned s = 0;
    for (int i = 0; i < 16; ++i) {
        local[i] = counts[tid * 16 + i];
        s += local[i];
    }
    tsum[tid] = s;
    __syncthreads();
    unsigned base = 0;
    for (int i = 0; i < tid; ++i) base += tsum[i];
    unsigned run = base;
    for (int i = 0; i < 16; ++i) {
        rowstart[tid * 16 + i] = run;
        run += local[i];
    }
}

// ---------------------------------------------------------------------------
// Pass 3: ordered compaction. One block per row; wave32 ballot + popcount
// prefix within wave, 8-wave LDS scan across waves -> exact row-major order.
// ---------------------------------------------------------------------------
__global__ __launch_bounds__(256) void nms_compact_kernel(
    const float* __restrict__ mask, const unsigned* __restrict__ rowstart,
    float* __restrict__ coords) {
    __shared__ unsigned wcnt[8];
    const int y = blockIdx.x;
    const int tid = threadIdx.x;
    const int lane = tid & 31;
    const int wid = tid >> 5;
    unsigned base = rowstart[y];
    const float* mrow = mask + (size_t)y * WW;

    for (int c = 0; c < 16; ++c) {
        const int x = c * 256 + tid;
        const bool pred = mrow[x] != 0.0f;
#ifdef HAVE_BALLOT_W32
        unsigned bal = __builtin_amdgcn_ballot_w32(pred);
#else
        unsigned bal = (unsigned)__ballot(pred);
#endif
        if (lane == 0) wcnt[wid] = (unsigned)__popc(bal);
        __syncthreads();
        unsigned wpre = 0, tot = 0;
        for (int w = 0; w < 8; ++w) {
            unsigned v = wcnt[w];
            tot += v;
            if (w < wid) wpre += v;
        }
        if (pred) {
            unsigned idx = base + wpre + (unsigned)__popc(bal & ((1u << lane) - 1u));
            if (idx < MAX_KPTS) {
                coords[idx] = (float)y;              // nz[2] = y row
                coords[MAX_KPTS + idx] = (float)x;   // nz[3] = x row
            }
        }
        base += tot;
        __syncthreads();  // protect wcnt for next chunk
    }
}

// ---------------------------------------------------------------------------
extern "C" void kernel_launch(void* const* d_in, const int* in_sizes, int n_in,
                              void* d_out, int out_size, void* d_ws, size_t ws_size,
                              hipStream_t stream) {
    const float* rel = (const float*)d_in[0];   // reliability
    const float* rep = (const float*)d_in[1];   // repeatability
    float* mask = (float*)d_out;                // [HH*WW] floats (0/1)
    float* coords = (float*)d_out + MASK_N;     // [2, MAX_KPTS] floats

    unsigned* counts = (unsigned*)d_ws;         // [4096]
    unsigned* rowstart = counts + HH;           // [4096]

    nms_fill_kernel<<<(2 * MAX_KPTS + 255) / 256, 256, 0, stream>>>(coords, 2 * MAX_KPTS, -1.0f);
    nms_mask_kernel<<<HH, 256, 0, stream>>>(rel, rep, mask, counts);
    nms_scan_kernel<<<1, 256, 0, stream>>>(counts, rowstart);
    nms_compact_kernel<<<HH, 256, 0, stream>>>(mask, rowstart, coords);
}